// MotorLayer_38293928411790
// MI455X (gfx1250) — compile-verified
//
#include <hip/hip_runtime.h>

// ---------------------------------------------------------------------------
// out[j] = sum_i weight[j,i] * input[i, (j<3)?0:1],  K = 8388608, 6 outputs.
// Pass 1: HBM-saturating fp32 streaming reduction -> per-block partials in ws.
// Pass 2: single wave folds partials with chained V_WMMA_F32_16X16X4_F32.
// ---------------------------------------------------------------------------

typedef float  v2f __attribute__((ext_vector_type(2)));
typedef float  v4f __attribute__((ext_vector_type(4)));
typedef float  v8f __attribute__((ext_vector_type(8)));

#define K_TOTAL   8388608
#define NBLK      1024
#define NTHR      256
#define ROW_PAD   16          // padded partial row: 6 real + 10 zeros

__global__ __launch_bounds__(NTHR)
void motor_partial_kernel(const float* __restrict__ inp,
                          const float* __restrict__ wgt,
                          float* __restrict__ ws)
{
    const int  tid    = blockIdx.x * NTHR + threadIdx.x;
    const int  stride = NBLK * NTHR;            // in float4 units
    const int  q4     = K_TOTAL / 4;            // 2097152 float4 per weight row

    const v4f* in4 = (const v4f*)inp;           // [K,2]: float4 q covers i=2q,2q+1
    const v4f* w0  = (const v4f*)(wgt + 0ull * K_TOTAL);
    const v4f* w1  = (const v4f*)(wgt + 1ull * K_TOTAL);
    const v4f* w2  = (const v4f*)(wgt + 2ull * K_TOTAL);
    const v4f* w3  = (const v4f*)(wgt + 3ull * K_TOTAL);
    const v4f* w4  = (const v4f*)(wgt + 4ull * K_TOTAL);
    const v4f* w5  = (const v4f*)(wgt + 5ull * K_TOTAL);

    float a0 = 0.f, a1 = 0.f, a2 = 0.f, a3 = 0.f, a4 = 0.f, a5 = 0.f;

    for (int p = tid; p < q4; p += stride) {
        // x01 = (x[4p],y[4p], x[4p+1],y[4p+1]) ; x23 = same for 4p+2,4p+3
        v4f x01 = __builtin_nontemporal_load(&in4[2 * p]);
        v4f x23 = __builtin_nontemporal_load(&in4[2 * p + 1]);
        v4f b0  = __builtin_nontemporal_load(&w0[p]);
        v4f b1  = __builtin_nontemporal_load(&w1[p]);
        v4f b2  = __builtin_nontemporal_load(&w2[p]);
        v4f b3  = __builtin_nontemporal_load(&w3[p]);
        v4f b4  = __builtin_nontemporal_load(&w4[p]);
        v4f b5  = __builtin_nontemporal_load(&w5[p]);

        // rows 0..2 use column 0 of input (x01.x, x01.z, x23.x, x23.z)
        a0 += b0.x * x01.x + b0.y * x01.z + b0.z * x23.x + b0.w * x23.z;
        a1 += b1.x * x01.x + b1.y * x01.z + b1.z * x23.x + b1.w * x23.z;
        a2 += b2.x * x01.x + b2.y * x01.z + b2.z * x23.x + b2.w * x23.z;
        // rows 3..5 use column 1 of input (x01.y, x01.w, x23.y, x23.w)
        a3 += b3.x * x01.y + b3.y * x01.w + b3.z * x23.y + b3.w * x23.w;
        a4 += b4.x * x01.y + b4.y * x01.w + b4.z * x23.y + b4.w * x23.w;
        a5 += b5.x * x01.y + b5.y * x01.w + b5.z * x23.y + b5.w * x23.w;
    }

    // --- wave32 butterfly reduction (deterministic) ---
    #pragma unroll
    for (int off = 16; off > 0; off >>= 1) {
        a0 += __shfl_xor(a0, off, 32);
        a1 += __shfl_xor(a1, off, 32);
        a2 += __shfl_xor(a2, off, 32);
        a3 += __shfl_xor(a3, off, 32);
        a4 += __shfl_xor(a4, off, 32);
        a5 += __shfl_xor(a5, off, 32);
    }

    __shared__ float red[NTHR / 32][8];         // 8 waves x 6 sums (padded)
    const int wv   = threadIdx.x >> 5;
    const int lane = threadIdx.x & 31;
    if (lane == 0) {
        red[wv][0] = a0; red[wv][1] = a1; red[wv][2] = a2;
        red[wv][3] = a3; red[wv][4] = a4; red[wv][5] = a5;
    }
    __syncthreads();

    // threads 0..15 emit one padded partial row per block (fixed order)
    if (threadIdx.x < ROW_PAD) {
        const int j = threadIdx.x;
        float s = 0.f;
        if (j < 6) {
            #pragma unroll
            for (int w = 0; w < NTHR / 32; ++w) s += red[w][j];
        }
        ws[(size_t)blockIdx.x * ROW_PAD + j] = s;   // cols 6..15 written as 0
    }
}

// Pass 2: one wave. D = A(ones,16x4) x B(partials,4x16) + C, chained over all
// 1024 partial rows, 4 rows per WMMA. With A==1, D[0,n] = sum_k B[k,n], so the
// exact K-slot <-> (VGPR, lane-half) mapping of B is irrelevant.
__global__ __launch_bounds__(32)
void motor_wmma_reduce_kernel(const float* __restrict__ ws,
                              float* __restrict__ out)
{
    const int lane = threadIdx.x;     // 0..31, EXEC all-1s (required by WMMA)
    const int col  = lane & 15;       // N coordinate
    const int hi   = lane >> 4;       // which pair of partial rows this half feeds

    v2f a; a.x = 1.0f; a.y = 1.0f;    // A = all ones
    v8f c = {};                        // C accumulator starts at 0

    for (int g = 0; g < NBLK; g += 4) {
        const int r = g + hi * 2;
        v2f b;
        b.x = ws[(size_t)(r + 0) * ROW_PAD + col];
        b.y = ws[(size_t)(r + 1) * ROW_PAD + col];
        c = __builtin_amdgcn_wmma_f32_16x16x4_f32(
                /*neg_a=*/false, a, /*neg_b=*/false, b,
                /*c_mod=*/(short)0, c, /*reuse_a=*/false, /*reuse_b=*/false);
    }

    if (lane < 6) out[lane] = c[0];   // VGPR0, lanes 0..15 => D[M=0][N=lane]
}

extern "C" void kernel_launch(void* const* d_in, const int* in_sizes, int n_in,
                              void* d_out, int out_size, void* d_ws, size_t ws_size,
                              hipStream_t stream)
{
    const float* inp = (const float*)d_in[0];   // [8388608, 2] fp32
    const float* wgt = (const float*)d_in[1];   // [6, 8388608] fp32
    float*       out = (float*)d_out;           // [6] fp32
    float*       ws  = (float*)d_ws;            // needs NBLK*16*4 = 64 KB

    motor_partial_kernel<<<NBLK, NTHR, 0, stream>>>(inp, wgt, ws);
    motor_wmma_reduce_kernel<<<1, 32, 0, stream>>>(ws, out);
}